// OptimizedPoseLossV3_74560632258756
// MI455X (gfx1250) — compile-verified
//
#include <hip/hip_runtime.h>

typedef __attribute__((ext_vector_type(4))) float f4;
typedef __attribute__((ext_vector_type(2))) float v2f;
typedef __attribute__((ext_vector_type(8))) float v8f;

#define NBLK 1024
#define NTHR 256
#define MLOG2 10
#define MDIM 1024

// Kernel 1: bandwidth-bound streaming pass.
// Reads pred/gt as float4 (one token-pair per element), computes t/s squared-error
// sums with the inter-view mask, and writes one float4 partial per block to d_ws.
__global__ void pose_loss_partials(const f4* __restrict__ pred,
                                   const f4* __restrict__ gt,
                                   const int* __restrict__ Ms,
                                   int V, int total,
                                   f4* __restrict__ blockOut)
{
    // Prefix-sum view boundaries (V <= 4 in the reference); kept in SGPRs.
    int c0 = 0x7fffffff, c1 = 0x7fffffff, c2 = 0x7fffffff;
    int acc = 0;
    if (V > 1) { acc += Ms[0]; c0 = acc; }
    if (V > 2) { acc += Ms[1]; c1 = acc; }
    if (V > 3) { acc += Ms[2]; c2 = acc; }

    float t_tot = 0.f, s_tot = 0.f, t_int = 0.f, s_int = 0.f;
    const int stride = gridDim.x * blockDim.x;

    #pragma unroll 4
    for (int p = blockIdx.x * blockDim.x + threadIdx.x; p < total; p += stride) {
        f4 pv = __builtin_nontemporal_load(&pred[p]);
        f4 gv = __builtin_nontemporal_load(&gt[p]);
        float d0 = pv.x - gv.x;
        float d1 = pv.y - gv.y;
        float d2 = pv.z - gv.z;
        float d3 = pv.w - gv.w;
        float t = d0 * d0 + d1 * d1;
        float s = d2 * d2 + d3 * d3;

        int j = p & (MDIM - 1);
        int i = (p >> MLOG2) & (MDIM - 1);
        int vi = (i >= c0) + (i >= c1) + (i >= c2);
        int vj = (j >= c0) + (j >= c1) + (j >= c2);
        float m = (vi != vj) ? 1.f : 0.f;

        t_tot += t;
        s_tot += s;
        t_int += t * m;
        s_int += s * m;
    }

    // wave32 reduction
    for (int off = 16; off > 0; off >>= 1) {
        t_tot += __shfl_down(t_tot, off);
        s_tot += __shfl_down(s_tot, off);
        t_int += __shfl_down(t_int, off);
        s_int += __shfl_down(s_int, off);
    }

    __shared__ float sw[NTHR / 32][4];
    const int lane = threadIdx.x & 31;
    const int wave = threadIdx.x >> 5;
    if (lane == 0) {
        sw[wave][0] = t_tot; sw[wave][1] = s_tot;
        sw[wave][2] = t_int; sw[wave][3] = s_int;
    }
    __syncthreads();
    if (threadIdx.x == 0) {
        float r0 = 0.f, r1 = 0.f, r2 = 0.f, r3 = 0.f;
        const int nwaves = blockDim.x >> 5;
        for (int w = 0; w < nwaves; ++w) {
            r0 += sw[w][0]; r1 += sw[w][1]; r2 += sw[w][2]; r3 += sw[w][3];
        }
        f4 o; o.x = r0; o.y = r1; o.z = r2; o.w = r3;
        blockOut[blockIdx.x] = o;
    }
}

// Kernel 2: single-block finalize. Reduces the NBLK x 4 partials; the 32-lane
// cross-lane sum uses V_WMMA_F32_16X16X4_F32 with A = ones(16x4) so that
// D[m][n] = sum_k B[k][n] — i.e. one matrix op folds 32 lanes into 16 columns
// per channel — then 4 shfl_xor steps finish it. Thread 0 applies the counts
// and alpha weights and writes the 7 outputs.
__global__ void pose_loss_finalize(const f4* __restrict__ blockOut, int nb,
                                   const int* __restrict__ Ms, int V,
                                   int B, float* __restrict__ out)
{
    float a[4] = {0.f, 0.f, 0.f, 0.f};
    for (int r = threadIdx.x; r < nb; r += blockDim.x) {
        f4 v = blockOut[r];
        a[0] += v.x; a[1] += v.y; a[2] += v.z; a[3] += v.w;
    }

    // EXEC is all-ones here (full 256-thread block, no divergence yet).
    v2f Aones; Aones.x = 1.f; Aones.y = 1.f;
    v8f Cz = {};
    float red[4];
    #pragma unroll
    for (int c = 0; c < 4; ++c) {
        v2f Bv; Bv.x = a[c]; Bv.y = 0.f;
        // D[m][n] = sum over K rows of B column n = a_c[lane n] + a_c[lane n+16]
        v8f D = __builtin_amdgcn_wmma_f32_16x16x4_f32(
            /*neg_a=*/false, Aones, /*neg_b=*/false, Bv,
            /*c_mod=*/(short)0, Cz, /*reuse_a=*/false, /*reuse_b=*/false);
        float r = D[0];                 // every lane holds colsum v[lane & 15]
        for (int off = 8; off > 0; off >>= 1) r += __shfl_xor(r, off);
        red[c] = r;                     // full 32-lane channel sum
    }

    __shared__ float sw[32][4];
    const int lane = threadIdx.x & 31;
    const int wave = threadIdx.x >> 5;
    if (lane == 0) {
        sw[wave][0] = red[0]; sw[wave][1] = red[1];
        sw[wave][2] = red[2]; sw[wave][3] = red[3];
    }
    __syncthreads();

    if (threadIdx.x == 0) {
        float t_tot = 0.f, s_tot = 0.f, t_int = 0.f, s_int = 0.f;
        const int nwaves = blockDim.x >> 5;
        for (int w = 0; w < nwaves; ++w) {
            t_tot += sw[w][0]; s_tot += sw[w][1];
            t_int += sw[w][2]; s_int += sw[w][3];
        }
        float sum_ms_sq = 0.f;
        for (int k = 0; k < V; ++k) {
            float mk = (float)Ms[k];
            sum_ms_sq += mk * mk;
        }
        const float fM = (float)MDIM;
        const float diag_count    = sum_ms_sq * (float)B;
        const float offdiag_count = (fM * fM - sum_ms_sq) * (float)B;

        const float intra_t = t_tot - t_int;
        const float intra_s = s_tot - s_int;

        const float loss_intra_t = intra_t / diag_count;
        const float loss_intra_s = intra_s / diag_count;
        const float loss_inter_t = t_int / offdiag_count;
        const float loss_inter_s = s_int / offdiag_count;

        const float ALPHA_T = 0.5f, ALPHA_S = 0.75f, ALPHA_TS = 0.5f;
        const float loss_t = ALPHA_T * loss_inter_t + (1.f - ALPHA_T) * loss_intra_t;
        const float loss_s = ALPHA_S * loss_inter_s + (1.f - ALPHA_S) * loss_intra_s;
        const float loss   = ALPHA_TS * loss_t + (1.f - ALPHA_TS) * loss_s;

        out[0] = loss_intra_t;
        out[1] = loss_inter_t;
        out[2] = loss_intra_s;
        out[3] = loss_inter_s;
        out[4] = loss_t;
        out[5] = loss_s;
        out[6] = loss;
    }
}

extern "C" void kernel_launch(void* const* d_in, const int* in_sizes, int n_in,
                              void* d_out, int out_size, void* d_ws, size_t ws_size,
                              hipStream_t stream) {
    const f4*  pred = (const f4*)d_in[0];
    const f4*  gt   = (const f4*)d_in[1];
    const int* Ms   = (const int*)d_in[2];
    const int  V    = in_sizes[2];
    const int  total = in_sizes[0] / 4;          // B*M*M token pairs
    const int  B     = total / (MDIM * MDIM);

    f4* blockOut = (f4*)d_ws;                    // NBLK float4 partials (16 KB)

    pose_loss_partials<<<NBLK, NTHR, 0, stream>>>(pred, gt, Ms, V, total, blockOut);
    pose_loss_finalize<<<1, 256, 0, stream>>>(blockOut, NBLK, Ms, V, B, (float*)d_out);
}